// SBMiniLayer_28587302323147
// MI455X (gfx1250) — compile-verified
//
#include <hip/hip_runtime.h>
#include <math.h>

// ---------------- problem constants ----------------
constexpr int B  = 128;
constexpr int N  = 2048;
constexpr int D  = 256;
constexpr int D2 = 512;
constexpr float INV_T = 1.0f / 0.3f;

// ---------------- output offsets (floats) ----------------
constexpr size_t O_MR   = 0;                               // memory_read  (B*D)
constexpr size_t O_KEYS = O_MR  + (size_t)B * D;           // updated_keys (B*N*D)
constexpr size_t O_VALS = O_KEYS + (size_t)B * N * D;      // updated_values
constexpr size_t O_STR  = O_VALS + (size_t)B * N * D;      // updated_strength (B*N)
constexpr size_t O_AGE  = O_STR + (size_t)B * N;           // updated_age (B*N)

// ---------------- workspace offsets (floats) ----------------
constexpr size_t WS_Q      = 0;                       // query raw -> normalized (B*D)
constexpr size_t WS_CK     = WS_Q    + (size_t)B * D; // cand_k raw -> tanh (B*D)
constexpr size_t WS_CV     = WS_CK   + (size_t)B * D; // cand_v raw -> tanh (B*D)
constexpr size_t WS_NCK    = WS_CV   + (size_t)B * D; // normalized cand_k (B*D)
constexpr size_t WS_GATES  = WS_NCK  + (size_t)B * D; // readmix[B], wstr[B], pers[B], mgpre[B]
constexpr size_t WS_SCORES = WS_GATES + (size_t)4 * B; // content scores (B*N)
constexpr size_t WS_SIM    = WS_SCORES + (size_t)B * N; // sim (B*N)
constexpr size_t WS_STATS  = WS_SIM  + (size_t)B * N;   // 16 floats per b
constexpr size_t WS_IDX    = WS_STATS + (size_t)16 * B; // 4 ints per b (aliased)
constexpr size_t WS_ACCC   = WS_IDX  + (size_t)4 * B;   // content read accum (B*D)
constexpr size_t WS_ACCP   = WS_ACCC + (size_t)B * D;   // persistence read accum (B*D)

typedef float v2f __attribute__((ext_vector_type(2)));
typedef float v4f __attribute__((ext_vector_type(4)));
typedef float v8f __attribute__((ext_vector_type(8)));

// Non-temporal 128-bit helpers on native clang vectors (streaming data
// >> 192MB L2: TH=NT keeps the stream from thrashing the cache).
__device__ __forceinline__ v4f ntload4(const v4f* p) {
  return __builtin_nontemporal_load(p);
}
__device__ __forceinline__ void ntstore4(v4f* p, v4f v) {
  __builtin_nontemporal_store(v, p);
}

// =====================================================================
// Kernel 1: 128x512x256 fp32 GEMM with V_WMMA_F32_16X16X4_F32.
// A = [signal | second] (row-major B x 512), W = (512 x 256) row-major.
// One wave computes one 16x16 C tile; K loop in steps of 4.
// A frag (16x4 f32): lanes 0-15 hold M=0..15 {K=k0,k0+1}, lanes 16-31 {K=k0+2,k0+3}.
// B frag (4x16 f32): vgpr0 = rows K=k0 (lanes 0-15) / k0+1 (lanes 16-31),
//                    vgpr1 = rows K=k0+2 / k0+3.
// C/D (16x16 f32): vgpr r -> (M=r, N=lane) lanes 0-15, (M=r+8) lanes 16-31.
// =====================================================================
__global__ __launch_bounds__(32) void wmma_gemm_kernel(
    const float* __restrict__ sig, const float* __restrict__ sec,
    const float* __restrict__ W, float* __restrict__ out)
{
  const int mT   = blockIdx.x;        // 0..7
  const int nT   = blockIdx.y;        // 0..15
  const int lane = threadIdx.x & 31;
  const int half = lane >> 4;         // 0 or 1
  const int l16  = lane & 15;

  const int m    = mT * 16 + l16;
  const int ncol = nT * 16 + l16;
  const float* arow0 = sig + (size_t)m * D;   // k < 256
  const float* arow1 = sec + (size_t)m * D;   // k >= 256

  v8f acc = {};
  for (int k0 = 0; k0 < D2; k0 += 4) {
    const int ka = k0 + half * 2;     // even, so ka and ka+1 in same half
    v2f a, b;
    if (ka < D) { a.x = arow0[ka];       a.y = arow0[ka + 1]; }
    else        { a.x = arow1[ka - D];   a.y = arow1[ka + 1 - D]; }
    const int kb = k0 + half;
    b.x = W[(size_t)kb * D + ncol];
    b.y = W[(size_t)(kb + 2) * D + ncol];
    acc = __builtin_amdgcn_wmma_f32_16x16x4_f32(
        /*neg_a=*/false, a, /*neg_b=*/false, b,
        /*c_mod=*/(short)0, acc, /*reuse_a=*/false, /*reuse_b=*/false);
  }
  for (int r = 0; r < 8; ++r) {
    const int mrow = mT * 16 + r + half * 8;
    out[(size_t)mrow * D + ncol] = acc[r];
  }
}

// =====================================================================
// Kernel 2: four scalar gates per batch row (512-length dot products).
// gates layout: [0,B)=read_mix  [B,2B)=write_strength  [2B,3B)=persistence
//               [3B,4B)=merge gate pre-activation
// =====================================================================
__global__ __launch_bounds__(256) void gates_kernel(
    const float* __restrict__ signal, const float* __restrict__ previous,
    const float* __restrict__ hidden,
    const float* __restrict__ Wrm, const float* __restrict__ brm,
    const float* __restrict__ Wwg, const float* __restrict__ bwg,
    const float* __restrict__ Wpg, const float* __restrict__ bpg,
    const float* __restrict__ Wmg, const float* __restrict__ bmg,
    float* __restrict__ gates)
{
  const int b = blockIdx.x, tid = threadIdx.x;
  __shared__ float red[256];
  const float* Ws[4] = {Wrm, Wwg, Wpg, Wmg};
  const float  bs[4] = {brm[0], bwg[0], bpg[0], bmg[0]};
  for (int g = 0; g < 4; ++g) {
    const float* W   = Ws[g];
    const float* sec = (g == 0) ? previous : hidden;  // joined_rp vs joined
    float s = 0.f;
    for (int k = tid; k < D2; k += 256) {
      const float x = (k < D) ? signal[(size_t)b * D + k]
                              : sec[(size_t)b * D + (k - D)];
      s += x * W[k];
    }
    red[tid] = s; __syncthreads();
    for (int o = 128; o; o >>= 1) {
      if (tid < o) red[tid] += red[tid + o];
      __syncthreads();
    }
    if (tid == 0) {
      const float v = red[0] + bs[g];
      gates[g * B + b] = (g < 3) ? 1.f / (1.f + __expf(-v)) : v;
    }
    __syncthreads();
  }
}

// =====================================================================
// Kernel 3: activations + row normalization (query and cand_k).
// =====================================================================
__global__ __launch_bounds__(256) void postproc_kernel(
    float* __restrict__ q, float* __restrict__ ck, float* __restrict__ cv,
    const float* __restrict__ bkp, const float* __restrict__ bvp,
    float* __restrict__ nck)
{
  const int b = blockIdx.x, d = threadIdx.x;
  __shared__ float red[256];
  const size_t idx = (size_t)b * D + d;
  const float qv  = q[idx];
  const float ckv = tanhf(ck[idx] + bkp[d]);
  const float cvv = tanhf(cv[idx] + bvp[d]);
  ck[idx] = ckv;
  cv[idx] = cvv;

  red[d] = qv * qv; __syncthreads();
  for (int o = 128; o; o >>= 1) { if (d < o) red[d] += red[d + o]; __syncthreads(); }
  const float qn = fmaxf(sqrtf(red[0]), 1e-6f);
  __syncthreads();

  red[d] = ckv * ckv; __syncthreads();
  for (int o = 128; o; o >>= 1) { if (d < o) red[d] += red[d + o]; __syncthreads(); }
  const float kn = fmaxf(sqrtf(red[0]), 1e-6f);

  q[idx]   = qv / qn;
  nck[idx] = ckv / kn;
}

// =====================================================================
// Kernel 4: single streaming pass over keys (256 MB):
//   scores[b,n] = (q . k) / max(||k||, eps),  sim[b,n] = (nck . k) / max(||k||, eps)
// One wave per n slot; 8 floats per lane (2x b128 NT loads);
// wave32 shfl_xor reductions.
// =====================================================================
__global__ __launch_bounds__(256) void scores_kernel(
    const float* __restrict__ keys, const float* __restrict__ query,
    const float* __restrict__ nck,
    float* __restrict__ scores, float* __restrict__ sim)
{
  const int b    = blockIdx.y;
  const int n0   = blockIdx.x * 64;
  const int wave = threadIdx.x >> 5;
  const int lane = threadIdx.x & 31;

  const v4f* qv = (const v4f*)(query + (size_t)b * D) + lane * 2;
  const v4f  q0 = qv[0], q1 = qv[1];
  const v4f* cp = (const v4f*)(nck + (size_t)b * D) + lane * 2;
  const v4f  c0 = cp[0], c1 = cp[1];

  for (int i = 0; i < 8; ++i) {
    const int n = n0 + wave * 8 + i;
    const v4f* kp = (const v4f*)(keys + ((size_t)b * N + n) * D) + lane * 2;
    const v4f k0 = ntload4(kp);
    const v4f k1 = ntload4(kp + 1);

    float ss = k0.x*k0.x + k0.y*k0.y + k0.z*k0.z + k0.w*k0.w
             + k1.x*k1.x + k1.y*k1.y + k1.z*k1.z + k1.w*k1.w;
    float sq = k0.x*q0.x + k0.y*q0.y + k0.z*q0.z + k0.w*q0.w
             + k1.x*q1.x + k1.y*q1.y + k1.z*q1.z + k1.w*q1.w;
    float sc = k0.x*c0.x + k0.y*c0.y + k0.z*c0.z + k0.w*c0.w
             + k1.x*c1.x + k1.y*c1.y + k1.z*c1.z + k1.w*c1.w;

    for (int o = 16; o; o >>= 1) {
      ss += __shfl_xor(ss, o, 32);
      sq += __shfl_xor(sq, o, 32);
      sc += __shfl_xor(sc, o, 32);
    }
    if (lane == 0) {
      const float nr = fmaxf(sqrtf(ss), 1e-6f);
      scores[(size_t)b * N + n] = sq / nr;
      sim[(size_t)b * N + n]    = sc / nr;
    }
  }
}

// =====================================================================
// Kernel 5: per-batch statistics + scalar target logic.
// stats[b*16 + 0..9] = {smax, 1/ssum, pmax, 1/psum, kmix, vmix, boostf, ow0, ow1, ow2}
// sidx[b*4 + 0..2]   = overwrite slot indices (-1 = unused)
// =====================================================================
__global__ __launch_bounds__(256) void stats_kernel(
    const float* __restrict__ scores, const float* __restrict__ sim,
    const float* __restrict__ strength, const float* __restrict__ age,
    const float* __restrict__ gW, const float* __restrict__ gP,
    const float* __restrict__ gM,
    float* __restrict__ stats, int* __restrict__ sidx)
{
  const int b = blockIdx.x, tid = threadIdx.x;
  __shared__ float red[256];
  __shared__ int   redi[256];
  __shared__ float t3v[768];
  __shared__ int   t3i[768];

  float smax = -1e30f, pmax = -1e30f;
  float v0 = -1e30f, v1 = -1e30f, v2 = -1e30f;
  int   j0 = -1, j1 = -1, j2 = -1;
  float rbest = -1e30f; int ribest = 0;

  for (int n = tid; n < N; n += 256) {
    const size_t ix = (size_t)b * N + n;
    const float s  = scores[ix];
    smax = fmaxf(smax, s);
    const float st = strength[ix], ag = age[ix];
    pmax = fmaxf(pmax, 2.4f * st + 0.6f * (1.f - ag));
    const float sm = sim[ix];
    if (sm > v0)      { v2 = v1; j2 = j1; v1 = v0; j1 = j0; v0 = sm; j0 = n; }
    else if (sm > v1) { v2 = v1; j2 = j1; v1 = sm; j1 = n; }
    else if (sm > v2) { v2 = sm; j2 = n; }
    const float rs = 1.2f * ag + (1.f - st) + 0.5f * (1.f - sm);
    if (rs > rbest) { rbest = rs; ribest = n; }
  }

  red[tid] = smax; __syncthreads();
  for (int o = 128; o; o >>= 1) { if (tid < o) red[tid] = fmaxf(red[tid], red[tid + o]); __syncthreads(); }
  const float smaxAll = red[0]; __syncthreads();

  red[tid] = pmax; __syncthreads();
  for (int o = 128; o; o >>= 1) { if (tid < o) red[tid] = fmaxf(red[tid], red[tid + o]); __syncthreads(); }
  const float pmaxAll = red[0]; __syncthreads();

  red[tid] = rbest; redi[tid] = ribest; __syncthreads();
  for (int o = 128; o; o >>= 1) {
    if (tid < o && red[tid + o] > red[tid]) { red[tid] = red[tid + o]; redi[tid] = redi[tid + o]; }
    __syncthreads();
  }
  const int replaceIdx = redi[0]; __syncthreads();

  t3v[tid] = v0; t3i[tid] = j0;
  t3v[256 + tid] = v1; t3i[256 + tid] = j1;
  t3v[512 + tid] = v2; t3i[512 + tid] = j2;
  __syncthreads();

  float se = 0.f, pe = 0.f;
  for (int n = tid; n < N; n += 256) {
    const size_t ix = (size_t)b * N + n;
    se += __expf((scores[ix] - smaxAll) * INV_T);
    const float st = strength[ix], ag = age[ix];
    pe += __expf((2.4f * st + 0.6f * (1.f - ag) - pmaxAll) * INV_T);
  }
  red[tid] = se; __syncthreads();
  for (int o = 128; o; o >>= 1) { if (tid < o) red[tid] += red[tid + o]; __syncthreads(); }
  const float ssum = red[0]; __syncthreads();
  red[tid] = pe; __syncthreads();
  for (int o = 128; o; o >>= 1) { if (tid < o) red[tid] += red[tid + o]; __syncthreads(); }
  const float psum = red[0];

  if (tid == 0) {
    float a0 = -1e30f, a1 = -1e30f, a2 = -1e30f;
    int   i0 = -1, i1 = -1, i2 = -1;
    for (int j = 0; j < 768; ++j) {
      const float v = t3v[j]; const int ii = t3i[j];
      if (ii < 0) continue;
      if (v > a0)      { a2 = a1; i2 = i1; a1 = a0; i1 = i0; a0 = v; i0 = ii; }
      else if (v > a1) { a2 = a1; i2 = i1; a1 = v; i1 = ii; }
      else if (v > a2) { a2 = v; i2 = ii; }
    }
    const float wsg  = gW[b];
    const float pers = gP[b];
    const float mergePref = 1.f / (1.f + __expf(-(gM[b] + 2.6f * a0)));
    const bool full    = (a0 > 0.78f) && (mergePref >= 0.55f);
    const bool multi   = full && (a1 > 0.68f);
    const bool partial = (!multi) && (a0 > 0.64f) && (a1 > 0.52f);
    const float novelty = fminf(fmaxf(1.f - a0, 0.f), 1.f);

    int   oi[3] = {-1, -1, -1};
    float ow[3] = {0.f, 0.f, 0.f};
    float scale;
    if (multi) {
      const float e0 = 1.f;
      const float e1 = __expf((a1 - a0) * INV_T);
      const float e2 = __expf((a2 - a0) * INV_T);
      const float inv = 1.f / (e0 + e1 + e2);
      oi[0] = i0; oi[1] = i1; oi[2] = i2;
      ow[0] = e0 * inv; ow[1] = e1 * inv; ow[2] = e2 * inv;
      scale = 0.16f + 0.52f * wsg;
    } else if (partial) {
      const float e0 = 1.f;
      const float e1 = __expf((a1 - a0) * INV_T);
      const float inv = 1.f / (e0 + e1);
      oi[0] = i0; oi[1] = i1;
      ow[0] = e0 * inv; ow[1] = e1 * inv;
      scale = 0.18f + 0.62f * wsg;
    } else if (full) {
      oi[0] = i0; ow[0] = 1.f;
      scale = 0.2f + 0.8f * wsg;
    } else {
      oi[0] = replaceIdx; ow[0] = 1.f;
      scale = 0.2f + 0.8f * wsg;
    }
    const float nf = 0.55f + 0.45f * novelty;
    ow[0] *= scale * nf; ow[1] *= scale * nf; ow[2] *= scale * nf;

    const bool mlike = full || partial || multi;
    float* st = stats + b * 16;
    st[0] = smaxAll; st[1] = 1.f / ssum;
    st[2] = pmaxAll; st[3] = 1.f / psum;
    st[4] = mlike ? 0.28f + 0.24f * pers : 0.78f + 0.16f * pers; // key_mix
    st[5] = mlike ? 0.42f + 0.28f * pers : 0.82f + 0.12f * pers; // value_mix
    st[6] = 0.45f + 0.35f * pers + 0.45f * novelty + 0.25f * wsg; // boost factor
    st[7] = ow[0]; st[8] = ow[1]; st[9] = ow[2];
    int* si = sidx + b * 4;
    si[0] = oi[0]; si[1] = oi[1]; si[2] = oi[2]; si[3] = 0;
  }
}

// =====================================================================
// Kernel 6: fused streaming update, all traffic in b128 NT ops.
// Block = (b, 128-slot chunk). Threads: 64 lanes x float4 across D,
// 4 groups across n. Reads keys+values, writes updated keys/values/
// strength/age; block-level shared reduction of read partial sums,
// then 2 float atomics per d.
// =====================================================================
__global__ __launch_bounds__(256) void update_kernel(
    const float* __restrict__ keys, const float* __restrict__ values,
    const float* __restrict__ strength, const float* __restrict__ age,
    const float* __restrict__ candk, const float* __restrict__ candv,
    const float* __restrict__ scores,
    const float* __restrict__ stats, const int* __restrict__ sidx,
    float* __restrict__ okeys, float* __restrict__ ovals,
    float* __restrict__ ostr, float* __restrict__ oage,
    float* __restrict__ accC, float* __restrict__ accP)
{
  const int b   = blockIdx.y;
  const int n0  = blockIdx.x * 128;
  const int tid = threadIdx.x;

  const float* st = stats + b * 16;
  const float smax = st[0], sinv = st[1], pmax = st[2], pinv = st[3];
  const float kmix = st[4], vmix = st[5], boostf = st[6];
  const float ow0 = st[7], ow1 = st[8], ow2 = st[9];
  const int i0 = sidx[b * 4], i1 = sidx[b * 4 + 1], i2 = sidx[b * 4 + 2];

  __shared__ float wc[128], wp[128], owsh[128];
  __shared__ float shc[1024], shp[1024];   // 4 n-groups x 256 d

  if (tid < 128) {
    const int n = n0 + tid;
    const size_t ix = (size_t)b * N + n;
    const float s = scores[ix];
    wc[tid] = __expf((s - smax) * INV_T) * sinv;
    const float stg = strength[ix], ag = age[ix];
    const float p = 2.4f * stg + 0.6f * (1.f - ag);
    wp[tid] = __expf((p - pmax) * INV_T) * pinv;
    const float ow = (n == i0) ? ow0 : (n == i1) ? ow1 : (n == i2) ? ow2 : 0.f;
    owsh[tid] = ow;
    ostr[ix] = fminf(fmaxf(stg * 0.995f + ow * boostf, 0.f), 1.f);
    oage[ix] = fminf(fmaxf((ag + 0.01f) * (1.f - ow), 0.f), 1.f);
  }
  __syncthreads();

  const int dq = (tid & 63) * 4;   // float4 offset in D
  const int jg = tid >> 6;         // n-group 0..3
  const v4f ck4 = *(const v4f*)(candk + (size_t)b * D + dq);
  const v4f cv4 = *(const v4f*)(candv + (size_t)b * D + dq);

  v4f ac = {0.f, 0.f, 0.f, 0.f};
  v4f ap = {0.f, 0.f, 0.f, 0.f};

  size_t base = ((size_t)b * N + n0 + jg) * D + dq;   // in floats
  for (int j = jg; j < 128; j += 4, base += (size_t)4 * D) {
    const v4f kv = ntload4((const v4f*)(keys + base));
    const v4f vv = ntload4((const v4f*)(values + base));
    const float w  = wc[j];
    const float p  = wp[j];
    const float ow = owsh[j];

    ac += w * vv;
    ap += p * vv;

    const float okm = ow * kmix;
    const float ovm = ow * vmix;
    const v4f uk = kv + okm * (ck4 - kv);
    const v4f uv = vv + ovm * (cv4 - vv);
    ntstore4((v4f*)(okeys + base), uk);
    ntstore4((v4f*)(ovals + base), uv);
  }

  // stage partial sums: group jg writes d-range [dq, dq+3]
  shc[jg * 256 + dq + 0] = ac.x; shc[jg * 256 + dq + 1] = ac.y;
  shc[jg * 256 + dq + 2] = ac.z; shc[jg * 256 + dq + 3] = ac.w;
  shp[jg * 256 + dq + 0] = ap.x; shp[jg * 256 + dq + 1] = ap.y;
  shp[jg * 256 + dq + 2] = ap.z; shp[jg * 256 + dq + 3] = ap.w;
  __syncthreads();

  // thread tid reduces d = tid over the 4 groups, single atomic per array
  const float c = shc[tid] + shc[256 + tid] + shc[512 + tid] + shc[768 + tid];
  const float q = shp[tid] + shp[256 + tid] + shp[512 + tid] + shp[768 + tid];
  atomicAdd(&accC[(size_t)b * D + tid], c);
  atomicAdd(&accP[(size_t)b * D + tid], q);
}

// =====================================================================
// Kernel 7: blend memory_read.
// =====================================================================
__global__ __launch_bounds__(256) void finalize_kernel(
    const float* __restrict__ accC, const float* __restrict__ accP,
    const float* __restrict__ readmix, float* __restrict__ out)
{
  const int b = blockIdx.x, d = threadIdx.x;
  const float m = readmix[b];
  const size_t ix = (size_t)b * D + d;
  out[ix] = m * accC[ix] + (1.f - m) * accP[ix];
}

// =====================================================================
extern "C" void kernel_launch(void* const* d_in, const int* in_sizes, int n_in,
                              void* d_out, int out_size, void* d_ws, size_t ws_size,
                              hipStream_t stream)
{
  const float* signal   = (const float*)d_in[0];
  const float* previous = (const float*)d_in[1];
  const float* hidden   = (const float*)d_in[2];
  const float* keys     = (const float*)d_in[3];
  const float* values   = (const float*)d_in[4];
  const float* strength = (const float*)d_in[5];
  const float* age      = (const float*)d_in[6];
  const float* W_rq     = (const float*)d_in[7];
  const float* W_rm     = (const float*)d_in[8];
  const float* b_rm     = (const float*)d_in[9];
  const float* W_kp     = (const float*)d_in[10];
  const float* b_kp     = (const float*)d_in[11];
  const float* W_vp     = (const float*)d_in[12];
  const float* b_vp     = (const float*)d_in[13];
  const float* W_wg     = (const float*)d_in[14];
  const float* b_wg     = (const float*)d_in[15];
  const float* W_pg     = (const float*)d_in[16];
  const float* b_pg     = (const float*)d_in[17];
  const float* W_mg     = (const float*)d_in[18];
  const float* b_mg     = (const float*)d_in[19];

  float* out = (float*)d_out;
  float* ws  = (float*)d_ws;

  // 1) WMMA projections: query raw, cand_k raw, cand_v raw
  wmma_gemm_kernel<<<dim3(8, 16), 32, 0, stream>>>(signal, previous, W_rq, ws + WS_Q);
  wmma_gemm_kernel<<<dim3(8, 16), 32, 0, stream>>>(signal, hidden,   W_kp, ws + WS_CK);
  wmma_gemm_kernel<<<dim3(8, 16), 32, 0, stream>>>(signal, hidden,   W_vp, ws + WS_CV);

  // 2) scalar gates
  gates_kernel<<<B, 256, 0, stream>>>(signal, previous, hidden,
                                      W_rm, b_rm, W_wg, b_wg, W_pg, b_pg, W_mg, b_mg,
                                      ws + WS_GATES);

  // 3) activations + normalization
  postproc_kernel<<<B, 256, 0, stream>>>(ws + WS_Q, ws + WS_CK, ws + WS_CV,
                                         b_kp, b_vp, ws + WS_NCK);

  // 4) streaming scores over keys
  scores_kernel<<<dim3(N / 64, B), 256, 0, stream>>>(keys, ws + WS_Q, ws + WS_NCK,
                                                     ws + WS_SCORES, ws + WS_SIM);

  // 5) per-batch stats + target logic
  stats_kernel<<<B, 256, 0, stream>>>(ws + WS_SCORES, ws + WS_SIM, strength, age,
                                      ws + WS_GATES + B, ws + WS_GATES + 2 * B,
                                      ws + WS_GATES + 3 * B,
                                      ws + WS_STATS, (int*)(ws + WS_IDX));

  // 6) zero read accumulators, then fused streaming update
  (void)hipMemsetAsync(ws + WS_ACCC, 0, (size_t)2 * B * D * sizeof(float), stream);
  update_kernel<<<dim3(N / 128, B), 256, 0, stream>>>(
      keys, values, strength, age,
      ws + WS_CK, ws + WS_CV, ws + WS_SCORES,
      ws + WS_STATS, (const int*)(ws + WS_IDX),
      out + O_KEYS, out + O_VALS, out + O_STR, out + O_AGE,
      ws + WS_ACCC, ws + WS_ACCP);

  // 7) blend memory_read
  finalize_kernel<<<B, 256, 0, stream>>>(ws + WS_ACCC, ws + WS_ACCP,
                                         ws + WS_GATES, out + O_MR);
}